// SoftXOR_78761110274679
// MI455X (gfx1250) — compile-verified
//
#include <hip/hip_runtime.h>

// XOR (dyadic) convolution of length-256 rows via separable 16x16 Hadamard
// transforms executed on CDNA5 WMMA (V_WMMA_F32_16X16X4_F32, exact fp32).
//
//   rv[b,:] = (1/256) * H * ( (H X H) .* (H Y H) ) * H,  X/Y = 16x16 reshape.
//
// One wave32 per batch row. H is generated in registers (entries +-1 from
// popcount), never loaded. Transposes between GEMM stages go through a
// padded, wave-private LDS tile.

typedef float v2f __attribute__((ext_vector_type(2)));
typedef float v8f __attribute__((ext_vector_type(8)));

#define WAVES_PER_BLOCK 8
#define RS 20 // padded LDS row stride in floats (16 data + 4 pad)

__device__ __forceinline__ float hval(int r, int c) {
  // 16x16 Hadamard: H[r][c] = (-1)^popcount(r & c)
  return (__builtin_popcount(r & c) & 1) ? -1.0f : 1.0f;
}

// B-operand fragment of H for K-chunk `chunk` (rows k0..k0+3 of H, all 16 cols).
// B 4x16 layout (wave32): vgpr v, lanes 0-15 -> K = k0+v, lanes 16-31 -> K = k0+2+v.
__device__ __forceinline__ v2f h_bfrag(int chunk, int lane) {
  const int n = lane & 15;
  const int k = chunk * 4 + ((lane >> 4) << 1);
  v2f hb;
  hb.x = hval(k, n);
  hb.y = hval(k + 1, n);
  return hb;
}

// D = M * H for the 16x16 tile stored row-major (stride RS) in `tile`.
// A 16x4 layout (wave32): lane = m + 16*(k>>1), vgpr = k&1.
__device__ __forceinline__ v8f mulH(const float* tile, int lane) {
  const int m = lane & 15;
  const int kh = (lane >> 4) << 1;
  v8f acc = {};
#pragma unroll
  for (int c = 0; c < 4; ++c) {
    v2f a;
    a.x = tile[m * RS + c * 4 + kh + 0];
    a.y = tile[m * RS + c * 4 + kh + 1];
    v2f hb = h_bfrag(c, lane);
    // 8 args: (neg_a, A, neg_b, B, c_mod, C, reuse_a, reuse_b)
    acc = __builtin_amdgcn_wmma_f32_16x16x4_f32(
        false, a, false, hb, (short)0, acc, false, false);
  }
  return acc;
}

// tile <- (tile * H)^T   (applying twice gives H * tile * H)
__device__ __forceinline__ void transform_T(float* tile, int lane) {
  v8f d = mulH(tile, lane);
  const int hi = lane >> 4;
  const int n = lane & 15;
  // D element (row = v + 8*hi, col = n) stored transposed.
#pragma unroll
  for (int v = 0; v < 8; ++v) {
    tile[n * RS + (v + 8 * hi)] = d[v];
  }
}

__global__ __launch_bounds__(WAVES_PER_BLOCK * 32)
void xor_conv_wht_kernel(const float* __restrict__ x, const float* __restrict__ y,
                         float* __restrict__ out, int B) {
  __shared__ float lds[WAVES_PER_BLOCK][2][16 * RS];
  const int lane = threadIdx.x & 31;
  const int wave = threadIdx.x >> 5;
  const int b = blockIdx.x * WAVES_PER_BLOCK + wave;
  if (b >= B) return; // wave-uniform exit; active waves keep EXEC all-ones

  float* tx = &lds[wave][0][0];
  float* ty = &lds[wave][1][0];

  // ---- cooperative load: lane handles 8 consecutive floats (2x b128) ----
  {
    const int row = lane >> 1;
    const int colb = (lane & 1) * 8;
    const float4* xp = reinterpret_cast<const float4*>(x + (size_t)b * 256) + lane * 2;
    const float4* yp = reinterpret_cast<const float4*>(y + (size_t)b * 256) + lane * 2;
    float4 x0 = xp[0], x1 = xp[1];
    float4 y0 = yp[0], y1 = yp[1];
    float* dx = tx + row * RS + colb;
    float* dy = ty + row * RS + colb;
    dx[0] = x0.x; dx[1] = x0.y; dx[2] = x0.z; dx[3] = x0.w;
    dx[4] = x1.x; dx[5] = x1.y; dx[6] = x1.z; dx[7] = x1.w;
    dy[0] = y0.x; dy[1] = y0.y; dy[2] = y0.z; dy[3] = y0.w;
    dy[4] = y1.x; dy[5] = y1.y; dy[6] = y1.z; dy[7] = y1.w;
  }

  // ---- forward 2D Hadamard of X and Y ----
  transform_T(tx, lane);
  transform_T(ty, lane);
  transform_T(tx, lane); // tx = H X H
  transform_T(ty, lane); // ty = H Y H

  // ---- elementwise product (spectral domain) ----
  {
    const int row = lane >> 1;
    const int colb = (lane & 1) * 8;
    float* px = tx + row * RS + colb;
    const float* py = ty + row * RS + colb;
#pragma unroll
    for (int t = 0; t < 8; ++t) px[t] *= py[t];
  }

  // ---- inverse 2D Hadamard; final stage stays in registers ----
  transform_T(tx, lane);
  v8f d = mulH(tx, lane); // d = H Z^T H = (H Z H)^T (C/D layout)

  const int hi = lane >> 4;
  const int n = lane & 15;
  float* ob = out + (size_t)b * 256;
#pragma unroll
  for (int v = 0; v < 8; ++v) {
    ob[(size_t)(v + 8 * hi) * 16 + n] = d[v] * (1.0f / 256.0f);
  }
}

extern "C" void kernel_launch(void* const* d_in, const int* in_sizes, int n_in,
                              void* d_out, int out_size, void* d_ws, size_t ws_size,
                              hipStream_t stream) {
  const float* x = (const float*)d_in[0];
  const float* y = (const float*)d_in[1];
  float* out = (float*)d_out;
  const int B = in_sizes[0] / 256;
  const int blocks = (B + WAVES_PER_BLOCK - 1) / WAVES_PER_BLOCK;
  hipLaunchKernelGGL(xor_conv_wht_kernel, dim3(blocks), dim3(WAVES_PER_BLOCK * 32),
                     0, stream, x, y, out, B);
}